// LocalizeAttention3D_11519102288201
// MI455X (gfx1250) — compile-verified
//
#include <hip/hip_runtime.h>
#include <cstdint>

#ifndef __has_builtin
#define __has_builtin(x) 0
#endif

#define AS1 __attribute__((address_space(1)))
#define AS3 __attribute__((address_space(3)))

typedef int v4i __attribute__((ext_vector_type(4)));

// Problem geometry (fixed by the reference harness).
constexpr int HV   = 24;             // H = W = D
constexpr int NVOX = HV * HV * HV;   // 13824 voxels
constexpr int DCH  = 16;             // feature dim
constexpr int KF   = 27;             // 3x3x3 neighbourhood
constexpr int BH   = 16;             // b*h = 2*8

// Tiling: one block handles (1 i-plane) x (JT j-rows) x (all 24 k) of one bh.
constexpr int JT       = 6;
constexpr int LDS_LJ   = JT + 2;                 // j halo
constexpr int LDS_VOX  = 3 * LDS_LJ * HV;        // 3 i-planes * 8 j * 24 k = 576
constexpr int OUT_F4   = JT * HV * KF * 4;       // float4 stores per block = 15552

__global__ __launch_bounds__(256)
void localize3d_async_kernel(const float* __restrict__ x, float* __restrict__ out) {
    // 576 voxels * 16 floats = 36 KB LDS tile (with halo, zero-padded).
    __shared__ __align__(16) float smem[LDS_VOX * DCH];

    const int tid = threadIdx.x;
    const int bh  = blockIdx.z;          // 0..15
    const int i   = blockIdx.y;          // 0..23
    const int j0  = blockIdx.x * JT;     // 0,6,12,18

    const float* xb = x + (size_t)bh * NVOX * DCH;

    // ---- Stage 1: fill LDS tile. In-volume slots come in via async copy
    // (ASYNCcnt path); out-of-volume halo slots are zero-filled via DS stores.
    // Addresses are disjoint, so no async/ds ordering hazard.
    for (int u = tid; u < LDS_VOX * 4; u += 256) {
        const int c4 = u & 3;        // which float4 of the 16-float vector
        const int s  = u >> 2;       // voxel slot 0..575
        const int sk = s % HV;
        const int t2 = s / HV;       // li*8 + lj
        const int lj = t2 & 7;
        const int li = t2 >> 3;
        const int si = i  + li - 1;
        const int sj = j0 + lj - 1;
        float* ldst = &smem[s * DCH + c4 * 4];
        if ((unsigned)si < (unsigned)HV && (unsigned)sj < (unsigned)HV) {
            const float* gsrc = xb + (((si * HV + sj) * HV + sk) * DCH + c4 * 4);
#if __has_builtin(__builtin_amdgcn_global_load_async_to_lds_b128)
            __builtin_amdgcn_global_load_async_to_lds_b128(
                (AS1 v4i*)gsrc, (AS3 v4i*)ldst, 0, 0);
#else
            unsigned loff = (unsigned)(uintptr_t)(AS3 void*)ldst;
            asm volatile("global_load_async_to_lds_b128 %0, %1, off"
                         :: "v"(loff), "v"(gsrc) : "memory");
#endif
        } else {
            *reinterpret_cast<float4*>(ldst) = make_float4(0.f, 0.f, 0.f, 0.f);
        }
    }

    // Drain this wave's async copies, then sync the workgroup so every wave
    // sees the fully-populated tile.
#if __has_builtin(__builtin_amdgcn_s_wait_asynccnt)
    __builtin_amdgcn_s_wait_asynccnt(0);
#else
    asm volatile("s_wait_asynccnt 0" ::: "memory");
#endif
    __syncthreads();

    // ---- Stage 2: emit the 27 shifted copies. Consecutive t -> consecutive
    // output float4s, so global_store_b128 is perfectly coalesced.
    float4* out4 = reinterpret_cast<float4*>(out);
    const size_t obase =
        ((size_t)bh * NVOX + (size_t)i * (HV * HV) + (size_t)j0 * HV) * (KF * 4);

    for (int t = tid; t < OUT_F4; t += 256) {
        const int c4 = t & 3;
        const int r  = t >> 2;       // voxel_in_tile * 27 + f
        const int f  = r % KF;       // filter index, order (fi, fj, fk)
        const int v  = r / KF;       // jj*24 + k
        const int k  = v % HV;
        const int jj = v / HV;
        const int fk = f % 3;
        const int ft = f / 3;
        const int fj = ft % 3;
        const int fi = ft / 3;
        const int sk = k + fk - 1;   // k halo handled by predicate (not in LDS)
        float4 val = make_float4(0.f, 0.f, 0.f, 0.f);
        if ((unsigned)sk < (unsigned)HV) {
            const int lidx = ((fi * LDS_LJ + (jj + fj)) * HV + sk) * DCH + c4 * 4;
            val = *reinterpret_cast<const float4*>(&smem[lidx]);
        }
        out4[obase + t] = val;
    }
}

extern "C" void kernel_launch(void* const* d_in, const int* in_sizes, int n_in,
                              void* d_out, int out_size, void* d_ws, size_t ws_size,
                              hipStream_t stream) {
    const float* x = (const float*)d_in[0];
    float* out = (float*)d_out;
    // grid: 4 j-tiles x 24 i-planes x 16 (b*h); 256 threads = 8 wave32s.
    dim3 grid(HV / JT, HV, BH);
    dim3 block(256);
    localize3d_async_kernel<<<grid, block, 0, stream>>>(x, out);
}